// PositionEmbeddingEncoder_54004918780159
// MI455X (gfx1250) — compile-verified
//
#include <hip/hip_runtime.h>
#include <stdint.h>

#ifndef __has_builtin
#define __has_builtin(x) 0
#endif

#if defined(__HIP_DEVICE_COMPILE__) && __has_builtin(__builtin_amdgcn_tensor_load_to_lds)
#define USE_TDM 1
#else
#define USE_TDM 0
#endif

// ---------------- constants (match reference) ----------------
namespace {
constexpr int   BLOCK        = 256;          // 8 wave32s
constexpr int   PTS_PER_ITER = BLOCK / 8;    // 8 threads per point -> 32 points/iter/block
constexpr int   MAX_BLOCKS   = 4096;
// floats per staged level table (levels 1..3): 8*2^(3d)
constexpr int   L1F = 64, L2F = 512, L3F = 4096;
constexpr int   LDSF = L1F + L2F + L3F;      // 4672 floats = 18,688 B LDS
}

typedef __attribute__((ext_vector_type(4))) unsigned int u32x4;
typedef __attribute__((ext_vector_type(8))) int          i32x8;
typedef __attribute__((ext_vector_type(4))) int          i32x4;
typedef __attribute__((ext_vector_type(4))) float        f32x4;   // native vec for b128 ops
typedef __attribute__((address_space(3)))   f32x4        as3_f32x4; // forces ds_load_b128

#if USE_TDM
// Issue one TDM 1-D tile load (nfloats contiguous f32) global -> LDS.
// D# packing per CDNA5 ISA ch.7/8 (group0 128b, group1 256b, groups 2/3 zero => <=2D).
__device__ __forceinline__ void tdm_load_1d(const float* gsrc, uint32_t lds_byte,
                                            uint32_t nfloats)
{
    uint64_t ga = (uint64_t)(uintptr_t)gsrc;
    u32x4 g0 = {
        1u,                                             // count=1, user descriptor
        lds_byte,                                       // lds_addr[31:0]
        (uint32_t)ga,                                   // global_addr[31:0]
        (uint32_t)((ga >> 32) & 0x01FFFFFFu) | 0x80000000u // global_addr[56:32] | type=2
    };
    i32x8 g1 = {
        (int)(2u << 16),                                // mask=0, data_size=2 (4 bytes)
        (int)((nfloats & 0xFFFFu) << 16),               // tensor_dim0[15:0] @ bits63:48
        (int)(((nfloats >> 16) & 0xFFFFu) | (1u << 16)),// tensor_dim0[31:16], tensor_dim1=1
        (int)((nfloats & 0xFFFFu) << 16),               // tile_dim0 @ bits127:112
        1,                                              // tile_dim1=1, tile_dim2=0
        (int)nfloats,                                   // tensor_dim0_stride[31:0]
        0, 0
    };
    i32x4 gz4 = {0, 0, 0, 0};
    i32x8 gz8 = {0, 0, 0, 0, 0, 0, 0, 0};
    __builtin_amdgcn_tensor_load_to_lds(g0, g1, gz4, gz4, gz8, 0);
}
#endif

__global__ __launch_bounds__(BLOCK)
void posenc_grid_gather(const float* __restrict__ x,
                        const float* __restrict__ w1, const float* __restrict__ w2,
                        const float* __restrict__ w3, const float* __restrict__ w4,
                        const float* __restrict__ w5, const float* __restrict__ w6,
                        const float* __restrict__ w7, const float* __restrict__ w8,
                        float* __restrict__ out, int n)
{
    __shared__ float tab[LDSF];   // levels 1..3 at float offsets 0 / 64 / 576

#if USE_TDM
    if (threadIdx.x < 32u) {      // wave 0 issues the DMA descriptors
        tdm_load_1d(w1, (uint32_t)(uintptr_t)&tab[0],   (uint32_t)L1F);
        tdm_load_1d(w2, (uint32_t)(uintptr_t)&tab[64],  (uint32_t)L2F);
        tdm_load_1d(w3, (uint32_t)(uintptr_t)&tab[576], (uint32_t)L3F);
#if __has_builtin(__builtin_amdgcn_s_wait_tensorcnt)
        __builtin_amdgcn_s_wait_tensorcnt(0);
#else
        asm volatile("s_wait_tensorcnt 0x0" ::: "memory");
#endif
    }
#else
    {   // cooperative fallback copy (b128 loads + ds_store_b128)
        f32x4*       d  = (f32x4*)tab;
        const f32x4* s1 = (const f32x4*)w1;   // 16 x f32x4
        const f32x4* s2 = (const f32x4*)w2;   // 128
        const f32x4* s3 = (const f32x4*)w3;   // 1024
        for (int i = threadIdx.x; i < 16;   i += BLOCK) d[i]        = s1[i];
        for (int i = threadIdx.x; i < 128;  i += BLOCK) d[16 + i]   = s2[i];
        for (int i = threadIdx.x; i < 1024; i += BLOCK) d[144 + i]  = s3[i];
    }
#endif
    __syncthreads();

    const int tid = threadIdx.x;
    const int lvl = tid & 7;          // 0..7  (depth = lvl+1)
    const int pin = tid >> 3;         // point slot within block, 0..31
    const int pstride = (int)gridDim.x * PTS_PER_ITER;

    // LDS byte offset of this level's staged table (generic shared addr low 32
    // bits == LDS offset per the aperture rules); used to build an explicit
    // addrspace(3) pointer so the gather stays a ds_load_b128 and cannot be
    // select-merged with the global branch into a flat_load.
    const uint32_t tab_base = (uint32_t)(uintptr_t)&tab[0];
    const uint32_t lds_lvl_byte =
        tab_base + ((lvl == 0) ? 0u : (lvl == 1) ? 64u * 4u : 576u * 4u);

    const float* gw = (lvl == 3) ? w4 : (lvl == 4) ? w5 :
                      (lvl == 5) ? w6 : (lvl == 6) ? w7 : w8;       // used for lvl>=3
    const int   gi = 2 << lvl;        // grid size 2^(lvl+1)
    const float gf = (float)gi;
    const float gm1 = gf - 1.0f;

    for (int p = (int)blockIdx.x * PTS_PER_ITER + pin; p < n; p += pstride) {
        const float* xp = x + 3u * (uint32_t)p;
        float fx = xp[0], fy = xp[1], fz = xp[2];

        // prefetch next point's coords (gfx1250 global_prefetch)
        if (p + pstride < n)
            __builtin_prefetch(x + 3u * (uint32_t)(p + pstride), 0, 0);

        // scale to [0,1), then per-level cell index with clamp
        float sx = (fx + 128.0f) * (1.0f / 256.0f);
        float sy = (fy + 128.0f) * (1.0f / 256.0f);
        float sz = (fz + 128.0f) * (1.0f / 256.0f);
        int ix = (int)fminf(fmaxf(floorf(sx * gf), 0.0f), gm1);
        int iy = (int)fminf(fmaxf(floorf(sy * gf), 0.0f), gm1);
        int iz = (int)fminf(fmaxf(floorf(sz * gf), 0.0f), gm1);
        int flat = ix + gi * (iy + gi * iz);

        f32x4 r0, r1;
        if (lvl < 3) {
            const as3_f32x4* lr =
                (const as3_f32x4*)(lds_lvl_byte + (uint32_t)flat * 32u);
            r0 = lr[0]; r1 = lr[1];                                  // ds_load_b128 x2
        } else {
            const f32x4* gr = (const f32x4*)(gw + (size_t)flat * 8u); // global_load_b128 x2
            r0 = gr[0]; r1 = gr[1];
        }

        // 32B per thread, 256B contiguous per point across 8 lanes; NT so the
        // 256MB output stream does not evict L2-resident tables (d7/d8 hit rate)
        f32x4* o = (f32x4*)(out + (size_t)p * 64u + (uint32_t)(lvl * 8));
        __builtin_nontemporal_store(r0, o);
        __builtin_nontemporal_store(r1, o + 1);
    }
}

extern "C" void kernel_launch(void* const* d_in, const int* in_sizes, int n_in,
                              void* d_out, int out_size, void* d_ws, size_t ws_size,
                              hipStream_t stream)
{
    const float* x  = (const float*)d_in[0];
    const float* w1 = (const float*)d_in[1];
    const float* w2 = (const float*)d_in[2];
    const float* w3 = (const float*)d_in[3];
    const float* w4 = (const float*)d_in[4];
    const float* w5 = (const float*)d_in[5];
    const float* w6 = (const float*)d_in[6];
    const float* w7 = (const float*)d_in[7];
    const float* w8 = (const float*)d_in[8];
    float* out = (float*)d_out;

    int n = in_sizes[0] / 3;                       // 1,000,000 points
    int need = (n + PTS_PER_ITER - 1) / PTS_PER_ITER;
    int blocks = need < MAX_BLOCKS ? need : MAX_BLOCKS;
    if (blocks < 1) blocks = 1;

    posenc_grid_gather<<<blocks, BLOCK, 0, stream>>>(x, w1, w2, w3, w4, w5, w6, w7, w8,
                                                     out, n);
}